// CombinedLoss_37056977830581
// MI455X (gfx1250) — compile-verified
//
#include <hip/hip_runtime.h>

// ---------------------------------------------------------------------------
// CombinedLoss for MI455X (gfx1250, wave32).
//
// Memory-bound: 128MB predictions + 32MB targets streamed once (~7us floor at
// 23.3 TB/s). All math in f32. WMMA (V_WMMA_F32_16X16X4_F32 with B=ones) is
// used as the wave-level 32-lane reduction: D = A x ones gives row sums of A,
// so one WMMA + 8 adds + one shfl_xor(16) reduces a full wave's partials in
// f32 on the matrix pipe.
// ---------------------------------------------------------------------------

typedef float v2f __attribute__((ext_vector_type(2)));
typedef float v8f __attribute__((ext_vector_type(8)));

// Problem constants (reference shapes are fixed).
#define BB 32
#define HH 1024
#define WW 1024
#define SS 256
#define ROWS_PER_BLOCK 8

// Reduce 32 per-lane partials to a single value (returned in all lanes).
// A(16x4): lane L, vgpr0 = partial, vgpr1 = 0  -> 64 A slots, 32 used.
// B(4x16) = all ones, C = 0. D[m,n] = rowsum_m(A), identical for every n.
// Per-lane sum of the 8 D vgprs: lanes 0-15 hold sum(rows 0-7), lanes 16-31
// hold sum(rows 8-15); shfl_xor(16) + add -> grand total in every lane.
// EXEC must be all ones at this point (callers keep the wave convergent).
__device__ __forceinline__ float wave_reduce_f32_wmma(float partial) {
    v2f a; a[0] = partial; a[1] = 0.0f;
    v2f b; b[0] = 1.0f;    b[1] = 1.0f;
    v8f c = {};
    c = __builtin_amdgcn_wmma_f32_16x16x4_f32(
        /*neg_a=*/false, a, /*neg_b=*/false, b,
        /*c_mod=*/(short)0, c, /*reuse_a=*/false, /*reuse_b=*/false);
    float s = c[0] + c[1] + c[2] + c[3] + c[4] + c[5] + c[6] + c[7];
    s += __shfl_xor(s, 16, 32);
    return s;
}

// Block reduction: WMMA wave reduce, then thread 0 folds the (<=8) wave sums
// from LDS and does ONE global f32 atomic per block.
__device__ __forceinline__ void block_reduce_atomic(float partial, float* dst,
                                                    float* sWave) {
    float w = wave_reduce_f32_wmma(partial);
    const int lane = threadIdx.x & 31;
    const int wid  = threadIdx.x >> 5;
    if (lane == 0) sWave[wid] = w;
    __syncthreads();
    if (threadIdx.x == 0) {
        float total = 0.0f;
        const int nw = blockDim.x >> 5;
        for (int i = 0; i < nw; ++i) total += sWave[i];
        atomicAdd(dst, total);
    }
}

__global__ void init_ws_kernel(float* ws) {
    ws[0] = 0.0f;
    ws[1] = 0.0f;
}

// ---------------------------------------------------------------------------
// Image loss: sum over all prediction pixels of (pred - bilinear(target))^2.
// Block = 8 rows x 256 threads; each thread handles one float4 chunk per row
// (W/4 == blockDim.x). Bilinear: src = dst*0.5 - 0.25, taps clamped to edge
// (identical to jax's renormalized edge weights for factor-2 upsampling).
// ---------------------------------------------------------------------------
__global__ void img_loss_kernel(const float* __restrict__ P,
                                const float* __restrict__ T,
                                float* __restrict__ ws) {
    __shared__ float sWave[32];
    const int Hh = HH >> 1, Wh = WW >> 1;

    float acc = 0.0f;
    const int row0 = blockIdx.x * ROWS_PER_BLOCK;
    const int x = threadIdx.x << 2;  // 4 pixels per thread per row

    #pragma unroll
    for (int rr = 0; rr < ROWS_PER_BLOCK; ++rr) {
        const int row = row0 + rr;           // global row in [0, B*H)
        const int y = row % HH;
        const int bb = row / HH;

        const float* prow = P + ((size_t)(bb * HH + y)) * WW + x;
        // Speculative prefetch of next row (gfx1250 global_prefetch_b8).
        __builtin_prefetch(prow + (size_t)WW, 0, 0);

        // Shared vertical bilinear weights for this row.
        const float sy = y * 0.5f - 0.25f;
        const float fy = floorf(sy);
        const int   y0 = (int)fy;
        const float wy = sy - fy;
        const int y0c = min(max(y0, 0), Hh - 1);
        const int y1c = min(y0 + 1, Hh - 1);  // y0+1 >= 0 always
        const float* trow0 = T + ((size_t)(bb * Hh + y0c)) * Wh;
        const float* trow1 = T + ((size_t)(bb * Hh + y1c)) * Wh;

        const float4 p4 = *(const float4*)prow;
        const float pv[4] = {p4.x, p4.y, p4.z, p4.w};

        #pragma unroll
        for (int j = 0; j < 4; ++j) {
            const int xx = x + j;
            const float sx = xx * 0.5f - 0.25f;
            const float fx = floorf(sx);
            const int   x0 = (int)fx;
            const float wx = sx - fx;
            const int x0c = min(max(x0, 0), Wh - 1);
            const int x1c = min(x0 + 1, Wh - 1);

            const float t00 = trow0[x0c], t01 = trow0[x1c];
            const float t10 = trow1[x0c], t11 = trow1[x1c];
            const float tv = (1.0f - wy) * ((1.0f - wx) * t00 + wx * t01)
                           +        wy  * ((1.0f - wx) * t10 + wx * t11);

            const float d = pv[j] - tv;
            acc = fmaf(d, d, acc);
        }
    }
    block_reduce_atomic(acc, &ws[0], sWave);
}

// ---------------------------------------------------------------------------
// Station loss: one thread per (batch, station); 3x3 clamped+masked box mean.
// Grid is exactly B*S threads, so every lane stays active (WMMA-safe).
// ---------------------------------------------------------------------------
__global__ void stn_loss_kernel(const float* __restrict__ P,
                                const int* __restrict__ pos,
                                const float* __restrict__ runoff,
                                float* __restrict__ ws) {
    __shared__ float sWave[32];
    const int idx = blockIdx.x * blockDim.x + threadIdx.x;  // [0, B*S)
    const int bb = idx / SS;
    const int px = pos[idx * 2 + 0];
    const int py = pos[idx * 2 + 1];
    const float* pb = P + (size_t)bb * HH * WW;

    float sum = 0.0f, cnt = 0.0f;
    #pragma unroll
    for (int dy = -1; dy <= 1; ++dy) {
        const int yy = py + dy;
        const bool vy = (yy >= 0) && (yy < HH);
        const int yc = min(max(yy, 0), HH - 1);
        #pragma unroll
        for (int dx = -1; dx <= 1; ++dx) {
            const int xx = px + dx;
            const bool vx = (xx >= 0) && (xx < WW);
            const int xc = min(max(xx, 0), WW - 1);
            const float m = (vy && vx) ? 1.0f : 0.0f;
            sum = fmaf(m, pb[(size_t)yc * WW + xc], sum);
            cnt += m;
        }
    }
    const float bm = sum / cnt;          // cnt >= 4 always (center in-range)
    const float d = bm - runoff[idx];
    block_reduce_atomic(d * d, &ws[1], sWave);
}

__global__ void finalize_kernel(const float* __restrict__ ws,
                                float* __restrict__ out) {
    const float img = ws[0] * (1.0f / ((float)BB * HH * WW));
    const float stn = ws[1] * (1.0f / ((float)BB * SS));
    out[0] = 1.0f * img + 0.5f * stn;    // IMAGE_W * img + STATION_W * stn
    out[1] = img;
    out[2] = stn;
}

extern "C" void kernel_launch(void* const* d_in, const int* in_sizes, int n_in,
                              void* d_out, int out_size, void* d_ws, size_t ws_size,
                              hipStream_t stream) {
    const float* preds   = (const float*)d_in[0];  // (32,1,1024,1024) f32
    const float* targets = (const float*)d_in[1];  // (32,1,512,512)   f32
    const int*   pos     = (const int*)d_in[2];    // (32,256,2)       i32
    const float* runoff  = (const float*)d_in[3];  // (32,256)         f32
    float* out = (float*)d_out;                    // 3 floats
    float* ws  = (float*)d_ws;                     // 2 f32 accumulators

    init_ws_kernel<<<1, 1, 0, stream>>>(ws);

    // Image loss: B*H rows / 8 rows-per-block = 4096 blocks of 256 threads.
    const int img_blocks = (BB * HH) / ROWS_PER_BLOCK;
    img_loss_kernel<<<img_blocks, 256, 0, stream>>>(preds, targets, ws);

    // Station loss: B*S = 8192 threads = 32 blocks of 256.
    stn_loss_kernel<<<(BB * SS) / 256, 256, 0, stream>>>(preds, pos, runoff, ws);

    finalize_kernel<<<1, 1, 0, stream>>>(ws, out);
}